// NodeAttentionEmbedding_71339406786695
// MI455X (gfx1250) — compile-verified
//
#include <hip/hip_runtime.h>
#include <hip/hip_bf16.h>
#include <cstdint>

// ---------------------------------------------------------------------------
// GAT-style layer, algebraically fused:
//   out = softmax_rows( 0.5 * ( aggcat @ Wcat.T  +  deg[n] * cc[c] ) )
// aggcat (N x 1024) = [ aggV0 | aggV1 | aggE0 | aggE1 | deg*X ]
// Wcat   (256 x 1024) = [ W3a0@W10 | W3a1@W11 | W3bb0@W20 | W3bb1@W21 | W3c0+W3c1 ]
// Attention scores computed directly from X / Ef via transformed vectors
// (h1, e2 never materialized). Final GEMM uses V_WMMA_F32_16X16X4_F32.
// ---------------------------------------------------------------------------

#define DIN_N  256
#define DIN_E  128
#define KCAT   1024
#define DMODEL 256

// vec-block layout (floats)
#define V_WVU 0      // 2*256  W1^T wVu
#define V_WVV 512    // 2*256  W1^T wVv
#define V_WEU 1024   // 2*256  W1^T wEu
#define V_WEV 1536   // 2*256  W1^T wEv
#define V_WEE 2048   // 2*128  W2^T wEe
#define V_BV  2304   // 2      score bias (V)
#define V_BE  2306   // 2      score bias (E)
#define V_CC  2308   // 256    per-column deg coefficient
#define V_TOTAL 2564

typedef float v2f __attribute__((ext_vector_type(2)));
typedef float v8f __attribute__((ext_vector_type(8)));

__device__ __forceinline__ float lrelu(float x) { return x > 0.f ? x : 0.01f * x; }

// order-preserving float -> uint key so we can use native global_atomic_max_u32
__device__ __forceinline__ unsigned ordkey(float f) {
    unsigned b = __float_as_uint(f);
    return b ^ ((unsigned)((int)b >> 31) | 0x80000000u);
}
__device__ __forceinline__ float ordval(unsigned u) {
    unsigned b = (u & 0x80000000u) ? (u ^ 0x80000000u) : ~u;
    return __uint_as_float(b);
}

__device__ __forceinline__ float wred_sum(float x) {
    for (int off = 16; off > 0; off >>= 1) x += __shfl_xor(x, off, 32);
    return x;
}
__device__ __forceinline__ float wred_max(float x) {
    for (int off = 16; off > 0; off >>= 1) x = fmaxf(x, __shfl_xor(x, off, 32));
    return x;
}

// ---------------------------------------------------------------------------
__global__ void k_zero(float* p, size_t n) {
    size_t i = (size_t)blockIdx.x * blockDim.x + threadIdx.x;
    size_t stride = (size_t)gridDim.x * blockDim.x;
    for (; i < n; i += stride) p[i] = 0.f;
}

// grid.x = 2 (head), 256 threads: transformed attention vectors + score biases
__global__ void k_attn_vecs(const float* __restrict__ W1w, const float* __restrict__ W1b,
                            const float* __restrict__ W2w, const float* __restrict__ W2b,
                            const float* __restrict__ aVw, const float* __restrict__ aVb,
                            const float* __restrict__ aEw, const float* __restrict__ aEb,
                            float* __restrict__ vecs) {
    const int i = blockIdx.x;       // head
    const int k = threadIdx.x;      // 0..255
    const float* W1 = W1w + (size_t)i * 256 * 256;
    const float* aV = aVw + (size_t)i * 512;
    const float* aE = aEw + (size_t)i * 640;
    float s_vu = 0.f, s_vv = 0.f, s_eu = 0.f, s_ev = 0.f;
    for (int c = 0; c < 256; ++c) {
        float w = W1[(size_t)c * 256 + k];
        s_vu += aV[c] * w;  s_vv += aV[256 + c] * w;
        s_eu += aE[c] * w;  s_ev += aE[256 + c] * w;
    }
    vecs[V_WVU + i * 256 + k] = s_vu;
    vecs[V_WVV + i * 256 + k] = s_vv;
    vecs[V_WEU + i * 256 + k] = s_eu;
    vecs[V_WEV + i * 256 + k] = s_ev;
    if (k < 128) {
        const float* W2 = W2w + (size_t)i * 128 * 128;
        float s = 0.f;
        for (int c = 0; c < 128; ++c) s += aE[512 + c] * W2[(size_t)c * 128 + k];
        vecs[V_WEE + i * 128 + k] = s;
    }
    if (k == 0) {
        float bv = aVb[i], be = aEb[i];
        for (int c = 0; c < 256; ++c) {
            float b = W1b[i * 256 + c];
            bv += b * (aV[c] + aV[256 + c]);
            be += b * (aE[c] + aE[256 + c]);
        }
        for (int c = 0; c < 128; ++c) be += W2b[i * 128 + c] * aE[512 + c];
        vecs[V_BV + i] = bv;
        vecs[V_BE + i] = be;
    }
}

// 1 block, 256 threads: cc[c] = sum_i (b1_i@W3a_i^T + b2_i@W3bb_i^T + W3b_i)[c]
__global__ void k_cc(const float* __restrict__ W3w, const float* __restrict__ W1b,
                     const float* __restrict__ W2b, const float* __restrict__ W3b,
                     float* __restrict__ vecs) {
    const int c = threadIdx.x;
    float s = 0.f;
    for (int i = 0; i < 2; ++i) {
        const float* w3 = W3w + (size_t)i * 256 * 640 + (size_t)c * 640;
        const float* b1 = W1b + i * 256;
        const float* b2 = W2b + i * 128;
        for (int k = 0; k < 256; ++k) s += b1[k] * w3[k];
        for (int k = 0; k < 128; ++k) s += b2[k] * w3[256 + k];
        s += W3b[i * 256 + c];
    }
    vecs[V_CC + c] = s;
}

// grid (256, 5), 256 threads: build Wcat (256 x 1024)
__global__ void k_wcat(const float* __restrict__ W3w, const float* __restrict__ W1w,
                       const float* __restrict__ W2w, float* __restrict__ Wcat) {
    const int c = blockIdx.x, seg = blockIdx.y, k = threadIdx.x;
    if (seg < 2) {                       // M1_i = W3a_i @ W1_i
        const int i = seg;
        const float* w3 = W3w + (size_t)i * 163840 + (size_t)c * 640;   // W3a row c
        const float* W1 = W1w + (size_t)i * 65536;
        float s = 0.f;
        for (int j = 0; j < 256; ++j) s += w3[j] * W1[(size_t)j * 256 + k];
        Wcat[(size_t)c * KCAT + i * 256 + k] = s;
    } else if (seg < 4) {                // M2_i = W3bb_i @ W2_i
        const int i = seg - 2;
        if (k < 128) {
            const float* w3 = W3w + (size_t)i * 163840 + (size_t)c * 640 + 256; // W3bb row c
            const float* W2 = W2w + (size_t)i * 16384;
            float s = 0.f;
            for (int j = 0; j < 128; ++j) s += w3[j] * W2[(size_t)j * 128 + k];
            Wcat[(size_t)c * KCAT + 512 + i * 128 + k] = s;
        }
    } else {                             // Wsum = W3c_0 + W3c_1
        Wcat[(size_t)c * KCAT + 768 + k] =
            W3w[(size_t)c * 640 + 384 + k] + W3w[163840 + (size_t)c * 640 + 384 + k];
    }
}

// wave per node: 8 dot products of X[n] against transformed vectors
__global__ __launch_bounds__(256) void k_pdots(const float* __restrict__ X,
                                               const float* __restrict__ vecs,
                                               float* __restrict__ pu, float* __restrict__ pv,
                                               int N) {
    __shared__ float sv[2048];
    for (int t = threadIdx.x; t < 2048; t += 256) sv[t] = vecs[t];
    __syncthreads();
    const int lane = threadIdx.x & 31, wave = threadIdx.x >> 5;
    const int n = blockIdx.x * 8 + wave;
    if (n >= N) return;
    const float* x = X + (size_t)n * DIN_N;
    float a0 = 0, a1 = 0, a2 = 0, a3 = 0, a4 = 0, a5 = 0, a6 = 0, a7 = 0;
    #pragma unroll
    for (int j = 0; j < 8; ++j) {
        int k = lane + 32 * j;
        float xv = x[k];
        a0 += xv * sv[V_WVU + k];        a1 += xv * sv[V_WVU + 256 + k];
        a2 += xv * sv[V_WVV + k];        a3 += xv * sv[V_WVV + 256 + k];
        a4 += xv * sv[V_WEU + k];        a5 += xv * sv[V_WEU + 256 + k];
        a6 += xv * sv[V_WEV + k];        a7 += xv * sv[V_WEV + 256 + k];
    }
    a0 = wred_sum(a0); a1 = wred_sum(a1); a2 = wred_sum(a2); a3 = wred_sum(a3);
    a4 = wred_sum(a4); a5 = wred_sum(a5); a6 = wred_sum(a6); a7 = wred_sum(a7);
    if (lane == 0) {
        ((float4*)pu)[n] = make_float4(a0, a1, a4, a5);   // {pVu0,pVu1,pEu0,pEu1}
        ((float4*)pv)[n] = make_float4(a2, a3, a6, a7);   // {pVv0,pVv1,pEv0,pEv1}
    }
}

// wave per edge: qE_i[e] = Ef[e] . (W2^T wEe_i)
__global__ __launch_bounds__(256) void k_qe(const float* __restrict__ Ef,
                                            const float* __restrict__ vecs,
                                            float* __restrict__ qE, int E) {
    __shared__ float sh[256];
    for (int t = threadIdx.x; t < 256; t += 256) sh[t] = vecs[V_WEE + t];
    __syncthreads();
    const int lane = threadIdx.x & 31, wave = threadIdx.x >> 5;
    const int e = blockIdx.x * 8 + wave;
    if (e >= E) return;
    const float* f = Ef + (size_t)e * DIN_E;
    float a0 = 0, a1 = 0;
    #pragma unroll
    for (int j = 0; j < 4; ++j) {
        int k = lane + 32 * j;
        float fv = f[k];
        a0 += fv * sh[k];
        a1 += fv * sh[128 + k];
    }
    a0 = wred_sum(a0); a1 = wred_sum(a1);
    if (lane == 0) { qE[2 * (size_t)e] = a0; qE[2 * (size_t)e + 1] = a1; }
}

// thread per edge: scores -> atomic segment max (ordered-uint) + degree
__global__ void k_scoremax(const int* __restrict__ EI,
                           const float* __restrict__ pu, const float* __restrict__ pv,
                           const float* __restrict__ qE, const float* __restrict__ vecs,
                           float* __restrict__ deg, unsigned* __restrict__ mV,
                           unsigned* __restrict__ mE, int N, int E) {
    const int e = blockIdx.x * blockDim.x + threadIdx.x;
    if (e >= E) return;
    const int u = EI[e], v = EI[E + e];
    const float4 PU = ((const float4*)pu)[u];
    const float4 PV = ((const float4*)pv)[v];
    const float2 q = ((const float2*)qE)[e];
    const float sV0 = lrelu(PU.x + PV.x + vecs[V_BV]);
    const float sV1 = lrelu(PU.y + PV.y + vecs[V_BV + 1]);
    const float sE0 = lrelu(PU.z + PV.z + q.x + vecs[V_BE]);
    const float sE1 = lrelu(PU.w + PV.w + q.y + vecs[V_BE + 1]);
    atomicAdd(&deg[u], 1.0f);
    atomicMax(&mV[u], ordkey(sV0));     atomicMax(&mV[N + u], ordkey(sV1));
    atomicMax(&mE[u], ordkey(sE0));     atomicMax(&mE[N + u], ordkey(sE1));
}

// thread per edge: exp(s - max) -> atomic denominators, store numerators
__global__ void k_expden(const int* __restrict__ EI,
                         const float* __restrict__ pu, const float* __restrict__ pv,
                         const float* __restrict__ qE, const float* __restrict__ vecs,
                         const unsigned* __restrict__ mV, const unsigned* __restrict__ mE,
                         float* __restrict__ denV, float* __restrict__ denE,
                         float* __restrict__ earr, int N, int E) {
    const int e = blockIdx.x * blockDim.x + threadIdx.x;
    if (e >= E) return;
    const int u = EI[e], v = EI[E + e];
    const float4 PU = ((const float4*)pu)[u];
    const float4 PV = ((const float4*)pv)[v];
    const float2 q = ((const float2*)qE)[e];
    const float sV0 = lrelu(PU.x + PV.x + vecs[V_BV]);
    const float sV1 = lrelu(PU.y + PV.y + vecs[V_BV + 1]);
    const float sE0 = lrelu(PU.z + PV.z + q.x + vecs[V_BE]);
    const float sE1 = lrelu(PU.w + PV.w + q.y + vecs[V_BE + 1]);
    const float eV0 = __expf(sV0 - ordval(mV[u]));
    const float eV1 = __expf(sV1 - ordval(mV[N + u]));
    const float eE0 = __expf(sE0 - ordval(mE[u]));
    const float eE1 = __expf(sE1 - ordval(mE[N + u]));
    atomicAdd(&denV[u], eV0);       atomicAdd(&denV[N + u], eV1);
    atomicAdd(&denE[u], eE0);       atomicAdd(&denE[N + u], eE1);
    ((float4*)earr)[e] = make_float4(eV0, eV1, eE0, eE1);
}

// wave per edge: segment-weighted scatter into aggcat (f32 atomics)
__global__ __launch_bounds__(256) void k_agg(const int* __restrict__ EI,
                                             const float* __restrict__ X,
                                             const float* __restrict__ Ef,
                                             const float* __restrict__ earr,
                                             const float* __restrict__ denV,
                                             const float* __restrict__ denE,
                                             float* __restrict__ agg, int N, int E) {
    const int lane = threadIdx.x & 31, wave = threadIdx.x >> 5;
    const int e = blockIdx.x * 8 + wave;
    if (e >= E) return;
    const int u = EI[e], v = EI[E + e];
    const float4 ee = ((const float4*)earr)[e];
    const float aV0 = ee.x / denV[u];
    const float aV1 = ee.y / denV[N + u];
    const float aE0 = ee.z / denE[u];
    const float aE1 = ee.w / denE[N + u];
    const float* x = X + (size_t)v * DIN_N;
    float* arow = agg + (size_t)u * KCAT;
    #pragma unroll
    for (int j = 0; j < 8; ++j) {
        int k = lane + 32 * j;
        float xv = x[k];
        atomicAdd(arow + k, aV0 * xv);
        atomicAdd(arow + 256 + k, aV1 * xv);
    }
    const float* f = Ef + (size_t)e * DIN_E;
    #pragma unroll
    for (int j = 0; j < 4; ++j) {
        int k = lane + 32 * j;
        float fv = f[k];
        atomicAdd(arow + 512 + k, aE0 * fv);
        atomicAdd(arow + 640 + k, aE1 * fv);
    }
}

// block per node: aggcat[n][768+k] = deg[n]*X[n][k]
__global__ void k_degx(const float* __restrict__ X, const float* __restrict__ deg,
                       float* __restrict__ agg, int N) {
    const int n = blockIdx.x, k = threadIdx.x;
    agg[(size_t)n * KCAT + 768 + k] = deg[n] * X[(size_t)n * DIN_N + k];
}

// ---------------------------------------------------------------------------
// Fused output GEMM: out = 0.5*( aggcat(Nx1024) @ Wcat(256x1024)^T + deg*cc )
// 256 threads = 8 waves: 2 (M) x 4 (N); each wave -> 16x64 strip via
// V_WMMA_F32_16X16X4_F32, A fragment reused across the 4 N-tiles.
// ---------------------------------------------------------------------------
__global__ __launch_bounds__(256) void k_gemm_out(const float* __restrict__ A,
                                                  const float* __restrict__ W,
                                                  const float* __restrict__ vecs,
                                                  const float* __restrict__ deg,
                                                  float* __restrict__ out, int M) {
    const int lane = threadIdx.x & 31;
    const int wave = threadIdx.x >> 5;
    const int wm = wave & 1;          // M sub-tile in block
    const int wn = wave >> 1;         // N strip (0..3) of 64 cols
    const int lm = lane & 15;
    const int kh = lane >> 4;         // half-wave -> K pair select
    const int m0 = blockIdx.x * 32 + wm * 16;
    int arow = m0 + lm; if (arow > M - 1) arow = M - 1;   // branchless-safe clamp
    const float* ap = A + (size_t)arow * KCAT + 2 * kh;
    const float* wp = W + (size_t)(wn * 64 + lm) * KCAT + 2 * kh;
    v8f acc0 = {}, acc1 = {}, acc2 = {}, acc3 = {};
    #pragma unroll 4
    for (int k0 = 0; k0 < KCAT; k0 += 4) {
        v2f a  = *(const v2f*)(ap + k0);
        v2f b0 = *(const v2f*)(wp + k0);
        v2f b1 = *(const v2f*)(wp + 16 * KCAT + k0);
        v2f b2 = *(const v2f*)(wp + 32 * KCAT + k0);
        v2f b3 = *(const v2f*)(wp + 48 * KCAT + k0);
        acc0 = __builtin_amdgcn_wmma_f32_16x16x4_f32(false, a, false, b0, (short)0, acc0, false, false);
        acc1 = __builtin_amdgcn_wmma_f32_16x16x4_f32(false, a, false, b1, (short)0, acc1, false, false);
        acc2 = __builtin_amdgcn_wmma_f32_16x16x4_f32(false, a, false, b2, (short)0, acc2, false, false);
        acc3 = __builtin_amdgcn_wmma_f32_16x16x4_f32(false, a, false, b3, (short)0, acc3, false, false);
    }
    const int n0 = wn * 64;
    const float* cc = vecs + V_CC;
    #pragma unroll
    for (int r = 0; r < 8; ++r) {
        const int row = m0 + r + 8 * kh;
        if (row < M) {
            const float dg = deg[row];
            float* orow = out + (size_t)row * DMODEL;
            orow[n0 +  0 + lm] = 0.5f * (acc0[r] + dg * cc[n0 +  0 + lm]);
            orow[n0 + 16 + lm] = 0.5f * (acc1[r] + dg * cc[n0 + 16 + lm]);
            orow[n0 + 32 + lm] = 0.5f * (acc2[r] + dg * cc[n0 + 32 + lm]);
            orow[n0 + 48 + lm] = 0.5f * (acc3[r] + dg * cc[n0 + 48 + lm]);
        }
    }
}

// wave per row: in-place row softmax over 256 columns
__global__ __launch_bounds__(256) void k_softmax(float* __restrict__ out, int N) {
    const int lane = threadIdx.x & 31, wave = threadIdx.x >> 5;
    const int n = blockIdx.x * 8 + wave;
    if (n >= N) return;
    float* row = out + (size_t)n * DMODEL;
    float v[8];
    float mx = -3.402823466e38f;
    #pragma unroll
    for (int j = 0; j < 8; ++j) { v[j] = row[lane + 32 * j]; mx = fmaxf(mx, v[j]); }
    mx = wred_max(mx);
    float s = 0.f;
    #pragma unroll
    for (int j = 0; j < 8; ++j) { v[j] = __expf(v[j] - mx); s += v[j]; }
    s = wred_sum(s);
    const float inv = 1.0f / s;
    #pragma unroll
    for (int j = 0; j < 8; ++j) row[lane + 32 * j] = v[j] * inv;
}

// ---------------------------------------------------------------------------
extern "C" void kernel_launch(void* const* d_in, const int* in_sizes, int n_in,
                              void* d_out, int out_size, void* d_ws, size_t ws_size,
                              hipStream_t stream) {
    const float* X   = (const float*)d_in[0];
    const float* Ef  = (const float*)d_in[1];
    const int*   EI  = (const int*)d_in[2];
    const float* W1w = (const float*)d_in[3];
    const float* W1b = (const float*)d_in[4];
    const float* W2w = (const float*)d_in[5];
    const float* W2b = (const float*)d_in[6];
    const float* W3w = (const float*)d_in[7];
    const float* W3b = (const float*)d_in[8];
    const float* aVw = (const float*)d_in[9];
    const float* aVb = (const float*)d_in[10];
    const float* aEw = (const float*)d_in[11];
    const float* aEb = (const float*)d_in[12];
    const int N = in_sizes[0] / DIN_N;   // 20000
    const int E = in_sizes[1] / DIN_E;   // 320000
    float* out = (float*)d_out;

    // workspace carve-up (floats); zero-initialized region first
    float* ws     = (float*)d_ws;
    float* agg    = ws;                                   // N*1024
    float* deg    = agg + (size_t)N * KCAT;               // N
    unsigned* mV  = (unsigned*)(deg + N);                 // 2N
    unsigned* mE  = mV + 2 * (size_t)N;                   // 2N
    float* denV   = (float*)(mE + 2 * (size_t)N);         // 2N
    float* denE   = denV + 2 * (size_t)N;                 // 2N
    const size_t zeroCount = (size_t)N * KCAT + (size_t)N * 9;
    float* pu     = denE + 2 * (size_t)N;                 // 4N
    float* pv     = pu + 4 * (size_t)N;                   // 4N
    float* qE     = pv + 4 * (size_t)N;                   // 2E
    float* earr   = qE + 2 * (size_t)E;                   // 4E
    float* Wcat   = earr + 4 * (size_t)E;                 // 256*1024
    float* vecs   = Wcat + (size_t)DMODEL * KCAT;         // V_TOTAL

    k_zero<<<4096, 256, 0, stream>>>(ws, zeroCount);
    k_attn_vecs<<<2, 256, 0, stream>>>(W1w, W1b, W2w, W2b, aVw, aVb, aEw, aEb, vecs);
    k_cc<<<1, 256, 0, stream>>>(W3w, W1b, W2b, W3b, vecs);
    k_wcat<<<dim3(256, 5), 256, 0, stream>>>(W3w, W1w, W2w, Wcat);
    k_pdots<<<(N + 7) / 8, 256, 0, stream>>>(X, vecs, pu, pv, N);
    k_qe<<<(E + 7) / 8, 256, 0, stream>>>(Ef, vecs, qE, E);
    k_scoremax<<<(E + 255) / 256, 256, 0, stream>>>(EI, pu, pv, qE, vecs, deg, mV, mE, N, E);
    k_expden<<<(E + 255) / 256, 256, 0, stream>>>(EI, pu, pv, qE, vecs, mV, mE, denV, denE, earr, N, E);
    k_agg<<<(E + 7) / 8, 256, 0, stream>>>(EI, X, Ef, earr, denV, denE, agg, N, E);
    k_degx<<<N, 256, 0, stream>>>(X, deg, agg, N);
    k_gemm_out<<<(N + 31) / 32, 256, 0, stream>>>(agg, Wcat, vecs, deg, out, N);
    k_softmax<<<(N + 7) / 8, 256, 0, stream>>>(out, N);
}